// SaltAndPepper_10471130267771
// MI455X (gfx1250) — compile-verified
//
#include <hip/hip_runtime.h>
#include <hip/hip_bf16.h>

// SaltAndPepper: out = (mask==0) ? 0 : (mask==1) ? 1 : image
// image (B,C,H,W) fp32, mask (B,1,H,W) int32 broadcast over C.
// Pure bandwidth-bound streaming kernel: ~235 MB moved -> ~10us floor @ 23.3 TB/s.

#define SB 32
#define SC 3
#define SH 512
#define SW 512

static constexpr int HW   = SH * SW;        // 262144
static constexpr int HW4  = HW / 4;         // 65536 float4 groups per plane
static constexpr int N4   = SB * HW4;       // 2,097,152 float4 groups of mask/pixels

typedef __attribute__((ext_vector_type(4))) float v4f;
typedef __attribute__((ext_vector_type(4))) int   v4i;

__device__ __forceinline__ float sp_sel(float img, int m) {
    // m==0 -> 0.0, m==1 -> 1.0, else img   (two v_cmp + two v_cndmask)
    return m == 0 ? 0.0f : (m == 1 ? 1.0f : img);
}

__global__ __launch_bounds__(256) void SaltAndPepper_10471130267771_kernel(
    const float* __restrict__ img,
    const int*   __restrict__ mask,
    float*       __restrict__ out)
{
    const int t = blockIdx.x * blockDim.x + threadIdx.x;   // float4-group id over (B,HW)
    if (t >= N4) return;

    const int b   = t / HW4;
    const int hw4 = t - b * HW4;

    // One 128-bit mask load covers 4 pixels, reused for all 3 channels (3x mask
    // traffic saving vs per-channel re-read).
    const v4i m = ((const v4i*)mask)[t];

#pragma unroll
    for (int c = 0; c < SC; ++c) {
        const int idx = (b * SC + c) * HW4 + hw4;          // float4 index into (B,C,HW)
        const v4f v = ((const v4f*)img)[idx];              // global_load_b128, RT
        v4f r;
        r.x = sp_sel(v.x, m.x);
        r.y = sp_sel(v.y, m.y);
        r.z = sp_sel(v.z, m.z);
        r.w = sp_sel(v.w, m.w);
        // Write-once output: non-temporal store keeps the read set L2-resident.
        __builtin_nontemporal_store(r, ((v4f*)out) + idx); // global_store_b128, NT
    }
}

extern "C" void kernel_launch(void* const* d_in, const int* in_sizes, int n_in,
                              void* d_out, int out_size, void* d_ws, size_t ws_size,
                              hipStream_t stream) {
    const float* img  = (const float*)d_in[0];
    const int*   mask = (const int*)d_in[1];
    float*       out  = (float*)d_out;

    const int threads = 256;                    // 8 wave32 per block
    const int blocks  = (N4 + threads - 1) / threads;   // exactly 8192
    SaltAndPepper_10471130267771_kernel<<<blocks, threads, 0, stream>>>(img, mask, out);
}